// QuantumMambaSSMCore_83863531421696
// MI455X (gfx1250) — compile-verified
//
#include <hip/hip_runtime.h>
#include <math.h>

#define NQ   10
#define DIM  1024        // 2^NQ amplitudes
#define BB   128         // batch
#define SS   512         // sequence length
#define DEG  3           // QSVT degree
#define PI_F 3.14159265358979323846f

typedef __attribute__((ext_vector_type(2))) float v2f;
typedef __attribute__((ext_vector_type(8))) float v8f;

// ---------------------------------------------------------------------------
// Kernel 1: WMMA prepass (V_WMMA_F32_16X16X4_F32, exact f32).
//   x_dbc[b,s,j] = sum_k angles[b,s,k] * W_x[j,k]   (M=65536, N=25->32, K=10->12)
//   dt_raw = x_dbc[:, :5]; C = x_dbc[:, 15:25]
//   dtang  = tanh(softplus(dt_raw . W_dt^T + b_dt)) * PI
// One wave32 per 16-row tile. All WMMA fragments come from zero-padded LDS
// (branch-free); epilogues go through LDS tiles and are written back as flat
// contiguous 160-float row copies (branch-free, coalesced).
// ---------------------------------------------------------------------------
__global__ __launch_bounds__(128) void qm_prepass(
    const float* __restrict__ angles, const float* __restrict__ Wx,
    const float* __restrict__ Wdt,    const float* __restrict__ bdt,
    float* __restrict__ Cws,          float* __restrict__ dtang)
{
  __shared__ float wxp[32 * 12];       // W_x^T padded [n][k]
  __shared__ float wdp[16 * 8];        // W_dt^T padded [n][k]
  __shared__ float atile[4][16 * 12];  // per-wave A tile, K padded to 12
  __shared__ float xtile[4][16 * 33];  // per-wave x_dbc tile (stride 33)
  __shared__ float dtt[4][16 * 17];    // per-wave dtang tile (stride 17)

  const int tid  = threadIdx.x;
  const int wave = tid >> 5;
  const int lane = tid & 31;
  const int half = lane >> 4;          // 0: lanes 0-15, 1: lanes 16-31
  const int mn   = lane & 15;
  const int R    = (blockIdx.x * 4 + wave) * 16;   // row base (b*S+s flat)

  // stage weights zero-padded (clamped address + value select, no branches)
  for (int idx = tid; idx < 32 * 12; idx += 128) {
    const int n = idx / 12, k = idx % 12;
    const float v = Wx[min(n, 24) * NQ + min(k, 9)];
    wxp[idx] = (n < 25 && k < NQ) ? v : 0.f;
  }
  for (int idx = tid; idx < 16 * 8; idx += 128) {
    const int n = idx / 8, k = idx % 8;
    const float v = Wdt[min(n, 9) * 5 + min(k, 4)];
    wdp[idx] = (n < NQ && k < 5) ? v : 0.f;
  }
  // stage A tile: 160 contiguous floats per wave, pad cols 10-11 with zero
  for (int idx = lane; idx < 160; idx += 32)
    atile[wave][(idx / 10) * 12 + (idx % 10)] = angles[R * NQ + idx];
  for (int idx = lane; idx < 16; idx += 32) {
    atile[wave][idx * 12 + 10] = 0.f;
    atile[wave][idx * 12 + 11] = 0.f;
  }
  __syncthreads();

  // ---- GEMM1 ----
  v8f acc0 = {}, acc1 = {};
#pragma unroll
  for (int t = 0; t < 3; ++t) {
    const int k0 = 4 * t + 2 * half;
    v2f a, b0, b1;
    a.x  = atile[wave][mn * 12 + k0];
    a.y  = atile[wave][mn * 12 + k0 + 1];
    b0.x = wxp[mn * 12 + k0];
    b0.y = wxp[mn * 12 + k0 + 1];
    b1.x = wxp[(16 + mn) * 12 + k0];
    b1.y = wxp[(16 + mn) * 12 + k0 + 1];
    acc0 = __builtin_amdgcn_wmma_f32_16x16x4_f32(false, a, false, b0, (short)0, acc0, false, false);
    acc1 = __builtin_amdgcn_wmma_f32_16x16x4_f32(false, a, false, b1, (short)0, acc1, false, false);
  }
  // dump fragments to LDS tile (unconditional), then zero K-pad cols 5..7
#pragma unroll
  for (int v = 0; v < 8; ++v) {
    const int m = v + 8 * half;
    xtile[wave][m * 33 + mn]      = acc0[v];
    xtile[wave][m * 33 + 16 + mn] = acc1[v];
  }
  for (int idx = lane; idx < 16; idx += 32) {
    xtile[wave][idx * 33 + 5] = 0.f;
    xtile[wave][idx * 33 + 6] = 0.f;
    xtile[wave][idx * 33 + 7] = 0.f;
  }
  __syncthreads();

  // C = x_dbc cols 15..24 : flat contiguous coalesced store
  for (int idx = lane; idx < 160; idx += 32)
    Cws[R * NQ + idx] = xtile[wave][(idx / 10) * 33 + 15 + (idx % 10)];

  // ---- GEMM2 (A = x_dbc cols 0..4, K padded with zeroed cols 5..7) ----
  v8f acc2 = {};
#pragma unroll
  for (int t = 0; t < 2; ++t) {
    const int k0 = 4 * t + 2 * half;
    v2f a, b;
    a.x = xtile[wave][mn * 33 + k0];
    a.y = xtile[wave][mn * 33 + k0 + 1];
    b.x = wdp[mn * 8 + k0];
    b.y = wdp[mn * 8 + k0 + 1];
    acc2 = __builtin_amdgcn_wmma_f32_16x16x4_f32(false, a, false, b, (short)0, acc2, false, false);
  }
  const float bn = bdt[min(mn, NQ - 1)];
#pragma unroll
  for (int v = 0; v < 8; ++v) {
    const int m = v + 8 * half;
    const float x  = acc2[v] + bn;
    const float sp = (x > 20.f) ? x : log1pf(__expf(x));   // softplus
    dtt[wave][m * 17 + mn] = tanhf(sp) * PI_F;             // cols >=10 unused
  }
  __syncthreads();
  for (int idx = lane; idx < 160; idx += 32)
    dtang[R * NQ + idx] = dtt[wave][(idx / 10) * 17 + (idx % 10)];
}

// ---------------------------------------------------------------------------
// Quad-gate helpers: one thread owns one quad (wires w, w+1), in-place.
// ---------------------------------------------------------------------------
__device__ __forceinline__ void apply_q4c(float* sre, float* sim, int tid,
                                          int w, const float* M /*32: re,im*/) {
  const unsigned st = 1u << w;
  const unsigned lo = (unsigned)tid & (st - 1u);
  const unsigned a  = (((unsigned)tid >> w) << (w + 2)) | lo;
  float r[4], im[4];
#pragma unroll
  for (int c = 0; c < 4; ++c) { r[c] = sre[a + c * st]; im[c] = sim[a + c * st]; }
#pragma unroll
  for (int rr = 0; rr < 4; ++rr) {
    float orr = 0.f, oi = 0.f;
#pragma unroll
    for (int cc = 0; cc < 4; ++cc) {
      const float mr = M[(rr * 4 + cc) * 2], mi = M[(rr * 4 + cc) * 2 + 1];
      orr += mr * r[cc] - mi * im[cc];
      oi  += mr * im[cc] + mi * r[cc];
    }
    sre[a + rr * st] = orr;  sim[a + rr * st] = oi;
  }
}

__device__ __forceinline__ void apply_q4r(float* sre, float* sim, int tid,
                                          int w, const float* M /*16 real*/) {
  const unsigned st = 1u << w;
  const unsigned lo = (unsigned)tid & (st - 1u);
  const unsigned a  = (((unsigned)tid >> w) << (w + 2)) | lo;
  float r[4], im[4];
#pragma unroll
  for (int c = 0; c < 4; ++c) { r[c] = sre[a + c * st]; im[c] = sim[a + c * st]; }
#pragma unroll
  for (int rr = 0; rr < 4; ++rr) {
    float orr = 0.f, oi = 0.f;
#pragma unroll
    for (int cc = 0; cc < 4; ++cc) {
      const float m = M[rr * 4 + cc];
      orr += m * r[cc];
      oi  += m * im[cc];
    }
    sre[a + rr * st] = orr;  sim[a + rr * st] = oi;
  }
}

// ---------------------------------------------------------------------------
// Kernel 2: sequential scan, one workgroup (256 threads, 8 wave32) per batch
// element. Amplitude-index bit i == wire i (consistent relabeling).
// Thread ownership: k = tid + j*256  =>  k bits 0..7 = tid, bits 8..9 = j.
// ---------------------------------------------------------------------------
__global__ __launch_bounds__(256) void qm_circuit(
    const float* __restrict__ angles, const float* __restrict__ Cws,
    const float* __restrict__ dtang,  const float* __restrict__ Dg,
    const float* __restrict__ cp,     const float* __restrict__ pc,
    float* __restrict__ out)
{
  __shared__ float sre[DIM], sim[DIM];
  __shared__ float um[20][8];                 // fused Rz*Ry*Rx per (layer,wire)
  __shared__ float um4[10][32];               // 4x4 complex quad gates (const)
  __shared__ float fm4[5][16];                // 4x4 real final-RY quads (per step)
  __shared__ unsigned short gq[DIM], gl[DIM]; // CNOT-chain gather maps
  __shared__ float hcur[NQ], dta[NQ], angL[NQ], cwL[NQ], Dl[NQ];
  __shared__ float ryh_c[NQ], ryh_s[NQ], fry_c[NQ], fry_s[NQ];
  __shared__ float plo[32], phi[32];          // product-state partial tables
  __shared__ float hz[DEG + 1][NQ];
  __shared__ float pcs[DEG + 1];
  __shared__ float zpart[8][NQ];

  const int tid  = threadIdx.x;
  const int b    = blockIdx.x;
  const int lane = tid & 31;
  const int wv   = tid >> 5;

  // ---- one-time init ----
  if (tid < 20) {   // fused per-wire layer unitaries U = Rz(g)*Ry(b)*Rx(a)
    const int k3 = tid * 3;
    float ca, sa, cb, sb, cg, sg;
    __sincosf(0.5f * cp[k3    ], &sa, &ca);
    __sincosf(0.5f * cp[k3 + 1], &sb, &cb);
    __sincosf(0.5f * cp[k3 + 2], &sg, &cg);
    const float m00r =  cb * ca, m00i =  sb * sa;
    const float m01r = -sb * ca, m01i = -cb * sa;
    const float m10r =  sb * ca, m10i = -cb * sa;
    const float m11r =  cb * ca, m11i = -sb * sa;
    float* u = um[tid];
    u[0] = cg*m00r + sg*m00i;  u[1] = cg*m00i - sg*m00r;   // row0 *= e^{-ig/2}
    u[2] = cg*m01r + sg*m01i;  u[3] = cg*m01i - sg*m01r;
    u[4] = cg*m10r - sg*m10i;  u[5] = cg*m10i + sg*m10r;   // row1 *= e^{+ig/2}
    u[6] = cg*m11r - sg*m11i;  u[7] = cg*m11i + sg*m11r;
  }
  if (tid < DEG + 1) pcs[tid] = pc[tid];
  if (tid < NQ) { Dl[tid] = Dg[tid]; hcur[tid] = 0.f; }
  for (int k = tid; k < DIM; k += 256) {
    // gather map G = Q(0,1) o ... o Q(8,9) o Q(9,0) (compose in reverse)
    unsigned x = (unsigned)k;
    x ^= (x >> 9) & 1u;                                    // cnot(9,0)
    for (int i = 8; i >= 0; --i) x ^= ((x >> i) & 1u) << (i + 1);
    gq[k] = (unsigned short)x;
    unsigned y = (unsigned)k;                              // without wrap
    for (int i = 8; i >= 0; --i) y ^= ((y >> i) & 1u) << (i + 1);
    gl[k] = (unsigned short)y;
  }
  __syncthreads();
  if (tid < 160) {  // Kronecker 4x4 quad gates: U_{w+1} (x) U_w
    const int pid = tid >> 4;          // L*5 + p
    const int e = tid & 15, rr = e >> 2, cc = e & 3;
    const int L = pid / 5, p = pid % 5;
    const float* uhi = um[L * NQ + 2 * p + 1];
    const float* ulo = um[L * NQ + 2 * p];
    const int hiI = (((rr >> 1) << 1) + (cc >> 1)) * 2;
    const int loI = (((rr & 1) << 1) + (cc & 1)) * 2;
    const float hr = uhi[hiI], hi = uhi[hiI + 1];
    const float lr = ulo[loI], li = ulo[loI + 1];
    um4[pid][e * 2]     = hr * lr - hi * li;
    um4[pid][e * 2 + 1] = hr * li + hi * lr;
  }
  __syncthreads();

  // ---- sequential scan over S steps ----
  for (int s = 0; s < SS; ++s) {
    const int base = (b * SS + s) * NQ;
    if (tid < NQ) {
      dta[tid]  = dtang[base + tid];
      angL[tid] = angles[base + tid];
      cwL[tid]  = Cws[base + tid];
    }
    __syncthreads();

    if (tid < NQ) {
      __sincosf(0.5f * hcur[tid], &ryh_s[tid], &ryh_c[tid]);
    } else if (tid < 50) {
      const int d = (tid - 10) / NQ, i = (tid - 10) % NQ;
      hz[d][i] = 0.5f * pcs[d] * dta[i] * PI_F;            // rz half-angles
    } else if (tid < 60) {
      const int i = tid - 50;
      __sincosf(0.5f * angL[i] * dta[i], &fry_s[i], &fry_c[i]);
    }
    __syncthreads();

    // product-state tables + final-RY quad tables
    if (tid < 32) {
      float v = 1.f;
#pragma unroll
      for (int i = 0; i < 5; ++i) v *= ((tid >> i) & 1) ? ryh_s[i] : ryh_c[i];
      plo[tid] = v;
    } else if (tid < 64) {
      const int m = tid - 32;
      float v = 1.f;
#pragma unroll
      for (int i = 0; i < 5; ++i) v *= ((m >> i) & 1) ? ryh_s[5 + i] : ryh_c[5 + i];
      phi[m] = v;
    } else if (tid < 144) {
      const int id = tid - 64;
      const int p = id >> 4, e = id & 15, rr = e >> 2, cc = e & 3;
      const int w0 = 2 * p, w1 = 2 * p + 1;
      const int b1 = rr >> 1, b0 = rr & 1, a1 = cc >> 1, a0 = cc & 1;
      const float eh = (a1 == b1) ? fry_c[w1] : (b1 ? fry_s[w1] : -fry_s[w1]);
      const float el = (a0 == b0) ? fry_c[w0] : (b0 ? fry_s[w0] : -fry_s[w0]);
      fm4[p][e] = eh * el;
    }
    __syncthreads();

    // build |psi> = (x)_i RY(h_i)|0> directly, fused with QSVT d=0 RZ phases.
    // signed half-angle sum: base over bits 0..7 (fixed per thread) +- h8 +- h9
    {
      float hh[NQ];
#pragma unroll
      for (int i = 0; i < NQ; ++i) hh[i] = hz[0][i];
      float bsum = 0.f;
#pragma unroll
      for (int i = 0; i < 8; ++i) bsum += ((tid >> i) & 1) ? hh[i] : -hh[i];
      const float pl = plo[tid & 31];
#pragma unroll
      for (int j = 0; j < 4; ++j) {
        const int k = tid + j * 256;
        const float prod = pl * phi[k >> 5];
        const float a = bsum + ((j & 1) ? hh[8] : -hh[8])
                             + ((j & 2) ? hh[9] : -hh[9]);
        float sn, cs; __sincosf(a, &sn, &cs);
        sre[k] = prod * cs;  sim[k] = prod * sn;
      }
    }
    __syncthreads();

    // QSVT rounds: permutation (all d) + fused diagonal RZ (d>=1)
    for (int d = 0; d <= DEG; ++d) {
      float tr[4], ti[4];
#pragma unroll
      for (int j = 0; j < 4; ++j) {
        const int k = tid + j * 256;  const int src = gq[k];
        tr[j] = sre[src];  ti[j] = sim[src];
      }
      __syncthreads();
#pragma unroll
      for (int j = 0; j < 4; ++j) {
        const int k = tid + j * 256;
        sre[k] = tr[j];  sim[k] = ti[j];
      }
      __syncthreads();
      if (d < DEG) {
        float hh[NQ];
#pragma unroll
        for (int i = 0; i < NQ; ++i) hh[i] = hz[d + 1][i];
        float bsum = 0.f;
#pragma unroll
        for (int i = 0; i < 8; ++i) bsum += ((tid >> i) & 1) ? hh[i] : -hh[i];
#pragma unroll
        for (int j = 0; j < 4; ++j) {
          const int k = tid + j * 256;
          const float a = bsum + ((j & 1) ? hh[8] : -hh[8])
                               + ((j & 2) ? hh[9] : -hh[9]);
          float sn, cs; __sincosf(a, &sn, &cs);
          const float r = sre[k], im = sim[k];
          sre[k] = cs * r - sn * im;  sim[k] = sn * r + cs * im;
        }
        __syncthreads();
      }
    }

    // two variational layers: 5 quad gates + CNOT-chain permutation each
    for (int L = 0; L < 2; ++L) {
#pragma unroll
      for (int p = 0; p < 5; ++p) {
        apply_q4c(sre, sim, tid, 2 * p, um4[L * 5 + p]);
        __syncthreads();
      }
      float tr[4], ti[4];
#pragma unroll
      for (int j = 0; j < 4; ++j) {
        const int k = tid + j * 256;  const int src = gl[k];
        tr[j] = sre[src];  ti[j] = sim[src];
      }
      __syncthreads();
#pragma unroll
      for (int j = 0; j < 4; ++j) {
        const int k = tid + j * 256;
        sre[k] = tr[j];  sim[k] = ti[j];
      }
      __syncthreads();
    }

    // final RY(x*dt): 5 real quad gates
#pragma unroll
    for (int p = 0; p < 5; ++p) {
      apply_q4r(sre, sim, tid, 2 * p, fm4[p]);
      __syncthreads();
    }

    // measurement <Z_i>: probs p_j for the 4 owned amplitudes; wires 0..7
    // have a fixed sign per thread, wires 8..9 are the j-combinations.
    float pj[4];
#pragma unroll
    for (int j = 0; j < 4; ++j) {
      const int k = tid + j * 256;
      const float r = sre[k], im = sim[k];
      pj[j] = r * r + im * im;
    }
    const float psum = (pj[0] + pj[1]) + (pj[2] + pj[3]);
    float z[NQ];
#pragma unroll
    for (int i = 0; i < 8; ++i) z[i] = ((tid >> i) & 1) ? -psum : psum;
    z[8] = (pj[0] + pj[2]) - (pj[1] + pj[3]);
    z[9] = (pj[0] + pj[1]) - (pj[2] + pj[3]);
#pragma unroll
    for (int off = 16; off > 0; off >>= 1)
#pragma unroll
      for (int i = 0; i < NQ; ++i) z[i] += __shfl_xor(z[i], off, 32);
    if (lane == 0)
#pragma unroll
      for (int i = 0; i < NQ; ++i) zpart[wv][i] = z[i];
    __syncthreads();
    if (tid < NQ) {
      float h2 = 0.f;
#pragma unroll
      for (int w = 0; w < 8; ++w) h2 += zpart[w][tid];
      hcur[tid] = h2;
      out[base + tid] = cwL[tid] * h2 + Dl[tid] * angL[tid];
    }
    __syncthreads();
  }
}

// ---------------------------------------------------------------------------
extern "C" void kernel_launch(void* const* d_in, const int* in_sizes, int n_in,
                              void* d_out, int out_size, void* d_ws, size_t ws_size,
                              hipStream_t stream) {
  const float* angles = (const float*)d_in[0];   // (128,512,10)
  const float* Wx     = (const float*)d_in[1];   // (25,10)
  const float* Wdt    = (const float*)d_in[2];   // (10,5)
  const float* bdt    = (const float*)d_in[3];   // (10)
  const float* pc     = (const float*)d_in[4];   // (4)
  const float* cp     = (const float*)d_in[5];   // (60)
  const float* Dg     = (const float*)d_in[6];   // (10)
  float* outp  = (float*)d_out;
  float* Cws   = (float*)d_ws;                          // B*S*NQ floats
  float* dtang = Cws + (size_t)BB * SS * NQ;            // B*S*NQ floats

  qm_prepass<<<(BB * SS) / 64, 128, 0, stream>>>(angles, Wx, Wdt, bdt, Cws, dtang);
  qm_circuit<<<BB, 256, 0, stream>>>(angles, Cws, dtang, Dg, cp, pc, outp);
}